// CrossMatchingLoss_23622320128213
// MI455X (gfx1250) — compile-verified
//
#include <hip/hip_runtime.h>
#include <math.h>

// Problem constants (match reference setup_inputs)
#define B_ 32
#define N_ 256
#define D_ 1024

typedef float v2f __attribute__((ext_vector_type(2)));
typedef float v8f __attribute__((ext_vector_type(8)));

// ---------------------------------------------------------------------------
// Kernel 1: reciprocal L2 norm per row: r = 1 / max(||x||, 1e-12)
// grid = (B_*N_, 2)  block = 256
// ---------------------------------------------------------------------------
__global__ void cml_rnorm_kernel(const float* __restrict__ s,
                                 const float* __restrict__ t,
                                 float* __restrict__ rs,
                                 float* __restrict__ rt) {
    const int row = blockIdx.x;
    const float* x = (blockIdx.y == 0 ? s : t) + (size_t)row * D_;
    float acc = 0.f;
    for (int k = threadIdx.x; k < D_; k += 256) {
        float v = x[k];
        acc += v * v;
    }
    __shared__ float sm[256];
    sm[threadIdx.x] = acc;
    __syncthreads();
    for (int off = 128; off > 0; off >>= 1) {
        if (threadIdx.x < off) sm[threadIdx.x] += sm[threadIdx.x + off];
        __syncthreads();
    }
    if (threadIdx.x == 0) {
        float nrm = sqrtf(sm[0]);
        float inv = 1.0f / fmaxf(nrm, 1e-12f);
        (blockIdx.y == 0 ? rs : rt)[row] = inv;
    }
}

// ---------------------------------------------------------------------------
// Kernel 2: sim[b] = s_n[b] @ t_n[b]^T via V_WMMA_F32_16X16X4_F32.
// One wave per 16x16 output tile; 8 waves (256 threads) per block.
// A fragment (16x4, f32, 2 VGPRs): lane<16 -> M=lane, K = k+0,k+1
//                                  lane>=16 -> M=lane-16, K = k+2,k+3
// B fragment (4x16) symmetric with N in place of M.
// C/D (16x16 f32, 8 VGPRs): VGPR r, lane l: M = r + 8*(l>>4), N = l&15.
// ---------------------------------------------------------------------------
__global__ void cml_simgemm_kernel(const float* __restrict__ S,
                                   const float* __restrict__ T,
                                   const float* __restrict__ rs,
                                   const float* __restrict__ rt,
                                   float* __restrict__ sim) {
    const int wid  = blockIdx.x * 8 + (threadIdx.x >> 5);  // global wave id
    const int b    = wid >> 8;                             // 256 tiles / batch
    const int tile = wid & 255;
    const int tm   = tile >> 4;
    const int tn   = tile & 15;
    const int lane = threadIdx.x & 31;
    const int r    = lane & 15;
    const int half = lane >> 4;

    const float* Ap = S + ((size_t)b * N_ + tm * 16 + r) * D_ + 2 * half;
    const float* Bp = T + ((size_t)b * N_ + tn * 16 + r) * D_ + 2 * half;

    v8f c = {0.f, 0.f, 0.f, 0.f, 0.f, 0.f, 0.f, 0.f};
    for (int k = 0; k < D_; k += 4) {
        v2f a  = *(const v2f*)(Ap + k);
        v2f bb = *(const v2f*)(Bp + k);
        c = __builtin_amdgcn_wmma_f32_16x16x4_f32(
                /*neg_a=*/false, a, /*neg_b=*/false, bb,
                /*c_mod=*/(short)0, c, /*reuse_a=*/false, /*reuse_b=*/false);
    }

    // scale by reciprocal norms and store
    const int   n   = r;                       // N within tile
    const float rtv = rt[b * N_ + tn * 16 + n];
    float* out = sim + (size_t)b * N_ * N_;
    for (int rr = 0; rr < 8; ++rr) {
        const int m   = rr + 8 * half;
        const float v = c[rr] * rs[b * N_ + tm * 16 + m] * rtv;
        out[(size_t)(tm * 16 + m) * N_ + tn * 16 + n] = v;
    }
}

// ---------------------------------------------------------------------------
// Kernel 3: Hungarian (Jonker-Volgenant) — ONE WAVE32 PER BATCH.
// No workgroup barriers: lockstep wave + in-order per-wave LDS ops +
// register shuffles. Each lane owns 8 contiguous columns (j-1 = lane*8+c),
// so each cost-row relaxation is two coalesced global_load_b128 per lane.
// float64 potentials to reproduce the reference's numpy-float64 matching.
// cost(i,j) = -sim[b][i][j].  Writes matched-sum (f64) to partial[b].
// ---------------------------------------------------------------------------
__global__ __launch_bounds__(32)
void cml_hungarian_kernel(const float* __restrict__ sim,
                          double* __restrict__ partial) {
    const int n    = N_;
    const int b    = blockIdx.x;
    const int lane = threadIdx.x;           // 0..31, single wave
    const float* Smat = sim + (size_t)b * n * n;
    const double INF = __builtin_inf();

    __shared__ double u_sh[N_ + 1];
    __shared__ int    p_sh[N_ + 1];
    __shared__ int    way_sh[N_ + 1];
    volatile double* u   = u_sh;
    volatile int*    p   = p_sh;
    volatile int*    way = way_sh;

    for (int k = lane; k <= n; k += 32) {
        u_sh[k] = 0.0;
        p_sh[k] = 0;
        way_sh[k] = 0;
    }
    __syncthreads();   // single-wave block: effectively a fence, near-free

    double vv[8];                    // v[j] for owned columns (registers)
    double minv[8];
    #pragma unroll
    for (int c = 0; c < 8; ++c) vv[c] = 0.0;

    for (int i = 1; i <= n; ++i) {
        if (lane == 0) p[0] = i;
        int j0 = 0;
        unsigned used_mask = 0u;     // bit c -> column j = lane*8 + c + 1
        #pragma unroll
        for (int c = 0; c < 8; ++c) minv[c] = INF;
        __builtin_amdgcn_wave_barrier();

        while (true) {
            // mark used[j0] (j0==0: the virtual column 0, p[0]=i; handled
            // below by the uniform u[i] += delta update)
            if (j0 > 0 && lane == ((j0 - 1) >> 3))
                used_mask |= 1u << ((j0 - 1) & 7);

            const int    i0   = p[j0];        // uniform LDS read
            const double u_i0 = u[i0];        // uniform LDS read

            // relaxation over this lane's 8 contiguous columns
            const float* row = Smat + (size_t)(i0 - 1) * n + lane * 8;
            const float4 f0 = *(const float4*)(row);
            const float4 f1 = *(const float4*)(row + 4);
            const float cv[8] = {f0.x, f0.y, f0.z, f0.w,
                                 f1.x, f1.y, f1.z, f1.w};
            double best  = INF;
            int    bestj = n + 1;
            #pragma unroll
            for (int c = 0; c < 8; ++c) {
                const int j = lane * 8 + c + 1;
                if (!((used_mask >> c) & 1u)) {
                    const double cur = -(double)cv[c] - u_i0 - vv[c];
                    if (cur < minv[c]) { minv[c] = cur; way[j] = j0; }
                    if (minv[c] < best) { best = minv[c]; bestj = j; }
                }
            }
            // wave-wide argmin, ties -> lowest column index (== np.argmin)
            #pragma unroll
            for (int off = 16; off > 0; off >>= 1) {
                const double ov = __shfl_xor(best, off, 32);
                const int    oj = __shfl_xor(bestj, off, 32);
                if (ov < best || (ov == best && oj < bestj)) {
                    best = ov;
                    bestj = oj;
                }
            }
            const double delta = best;   // uniform after butterfly
            const int    j1    = bestj;

            // potentials update (u rows of used columns are distinct ->
            // scattered LDS RMWs are race-free)
            if (lane == 0) u[i] += delta;            // virtual column 0
            #pragma unroll
            for (int c = 0; c < 8; ++c) {
                if ((used_mask >> c) & 1u) {
                    const int j = lane * 8 + c + 1;
                    u[p[j]] += delta;
                    vv[c]   -= delta;
                } else {
                    minv[c] -= delta;
                }
            }
            __builtin_amdgcn_wave_barrier();

            j0 = j1;
            if (p[j0] == 0) break;       // uniform LDS read
        }

        // augment along alternating tree (uniform serial chain walk)
        {
            int j = j0;
            while (j) {
                const int jn = way[j];
                const int pv = p[jn];
                if (lane == 0) p[j] = pv;
                j = jn;
            }
        }
        __builtin_amdgcn_wave_barrier();
    }

    // matched similarity: column j is matched to row p[j]-1
    double acc = 0.0;
    for (int c = 0; c < 8; ++c) {
        const int j = lane * 8 + c + 1;
        acc += (double)Smat[(size_t)(p[j] - 1) * n + (j - 1)];
    }
    #pragma unroll
    for (int off = 16; off > 0; off >>= 1) acc += __shfl_xor(acc, off, 32);
    if (lane == 0) partial[b] = acc;
}

// ---------------------------------------------------------------------------
// Kernel 4: deterministic final reduction: out = -sum(partial) / (B*N)
// ---------------------------------------------------------------------------
__global__ void cml_finalize_kernel(const double* __restrict__ partial,
                                    float* __restrict__ out) {
    if (threadIdx.x == 0 && blockIdx.x == 0) {
        double s = 0.0;
        for (int b = 0; b < B_; ++b) s += partial[b];
        out[0] = (float)(-s / (double)(B_ * N_));
    }
}

// ---------------------------------------------------------------------------
extern "C" void kernel_launch(void* const* d_in, const int* in_sizes, int n_in,
                              void* d_out, int out_size, void* d_ws, size_t ws_size,
                              hipStream_t stream) {
    (void)in_sizes; (void)n_in; (void)out_size; (void)ws_size;
    const float* s = (const float*)d_in[0];
    const float* t = (const float*)d_in[1];
    float* out = (float*)d_out;

    // workspace layout (floats): sim[B*N*N] | rs[B*N] | rt[B*N] | partial (f64)
    float*  sim     = (float*)d_ws;
    float*  rs      = sim + (size_t)B_ * N_ * N_;
    float*  rt      = rs + (size_t)B_ * N_;
    double* partial = (double*)(rt + (size_t)B_ * N_);  // 8-byte aligned offset

    // 1) reciprocal norms
    cml_rnorm_kernel<<<dim3(B_ * N_, 2), 256, 0, stream>>>(s, t, rs, rt);

    // 2) WMMA batched GEMM: 32 batches * 256 tiles = 8192 waves, 8 waves/block
    cml_simgemm_kernel<<<(B_ * 16 * 16) / 8, 256, 0, stream>>>(s, t, rs, rt, sim);

    // 3) Hungarian: one wave32 per batch, barrier-free inner loop
    cml_hungarian_kernel<<<B_, 32, 0, stream>>>(sim, partial);

    // 4) deterministic scalar finalize
    cml_finalize_kernel<<<1, 32, 0, stream>>>(partial, out);
}